// VectorQuantizerEMA_6571299963042
// MI455X (gfx1250) — compile-verified
//
#include <hip/hip_runtime.h>
#include <hip/hip_bf16.h>

typedef __attribute__((ext_vector_type(2))) float v2f;
typedef __attribute__((ext_vector_type(8))) float v8f;

#define VQ_N      131072      // B*H*W
#define VQ_D      64          // embed_dim == C
#define VQ_K      512         // num_embed
#define VQ_HW     4096        // H*W
#define OUT_LOSS  0
#define OUT_Q     1
#define OUT_PERP  (1 + 32*64*64*64)          // 8388609
#define OUT_IDX   (OUT_PERP + 1)             // 8388610

// ws[0] = SSE accumulator, ws[1..512] = code usage counts
__global__ void vq_init_ws(float* __restrict__ ws) {
    int t = threadIdx.x;
    if (t < VQ_K + 1) ws[t] = 0.0f;
}

__global__ __launch_bounds__(256)
void vq_main(const float* __restrict__ in, const float* __restrict__ ew,
             float* __restrict__ out, float* __restrict__ ws) {
    extern __shared__ float smem[];
    float* sm_code = smem;                 // 512*64 floats
    float* sm_en   = smem + VQ_K * VQ_D;   // 512 floats
    float* sm_sum  = sm_en + VQ_K;         // 1 float

    const int tid = threadIdx.x;

    // ---- stage codebook into LDS via CDNA5 async copies (ASYNCcnt path) ----
    // 32768 floats = 8192 x b128; 32 per thread. Generic LDS pointer's low 32
    // bits are the LDS byte offset (aperture rule), used as the VDST operand.
    {
        unsigned lds0 = (unsigned)(unsigned long long)(uintptr_t)sm_code;
        for (int i = tid * 4; i < VQ_K * VQ_D; i += 256 * 4) {
            unsigned long long ga = (unsigned long long)(uintptr_t)(ew + i);
            unsigned la = lds0 + (unsigned)(i * 4);
            asm volatile("global_load_async_to_lds_b128 %0, %1, off"
                         :: "v"(la), "v"(ga) : "memory");
        }
        asm volatile("s_wait_asynccnt 0x0" ::: "memory");
    }
    __syncthreads();

    // ---- per-code squared norms ----
    for (int k = tid; k < VQ_K; k += 256) {
        float s = 0.0f;
        #pragma unroll 8
        for (int c = 0; c < VQ_D; ++c) { float v = sm_code[k * VQ_D + c]; s += v * v; }
        sm_en[k] = s;
    }
    if (tid == 0) sm_sum[0] = 0.0f;
    __syncthreads();

    const int wave = tid >> 5;
    const int lane = tid & 31;
    const int l15  = lane & 15;
    const int hi   = lane >> 4;                   // 0: low K-half, 1: high K-half
    const int n0   = (blockIdx.x * 8 + wave) * 32; // 32 rows per wave (two 16-row tiles)
    const int b    = n0 >> 12;                     // / (H*W)
    const int h    = (n0 >> 6) & 63;
    const int w0   = n0 & 63;                      // 0 or 32

    // ---- load B fragments: two x tiles, 16 rows x 64 channels each ----
    // lane holds row (w0[+16] + l15), channels c = 4*kc + 2*hi + {0,1}
    const float* xb0 = in + (((size_t)(b * 64) * 64 + h) * 64 + w0 + l15);
    const float* xb1 = xb0 + 16;
    v2f xa0[16], xa1[16];
    #pragma unroll
    for (int kc = 0; kc < 16; ++kc) {
        size_t off = (size_t)(kc * 4 + hi * 2) * VQ_HW;
        xa0[kc].x = xb0[off]; xa0[kc].y = xb0[off + VQ_HW];
        xa1[kc].x = xb1[off]; xa1[kc].y = xb1[off + VQ_HW];
    }

    // ---- argmin over 512 codes via fp32 WMMA (dist = ||e||^2 - 2 x.e) ----
    // Each A-fragment (code tile) feeds two WMMAs -> half the DS bytes per FLOP.
    float mv0 = 3.402823466e38f, mv1 = 3.402823466e38f;
    int   mi0 = 0, mi1 = 0;
    #pragma unroll 1
    for (int t = 0; t < 32; ++t) {
        const int code0 = t * 16;
        v8f a0 = {0.f, 0.f, 0.f, 0.f, 0.f, 0.f, 0.f, 0.f};
        v8f a1 = {0.f, 0.f, 0.f, 0.f, 0.f, 0.f, 0.f, 0.f};
        // A fragment: lane holds code (code0 + l15), channels 4*kc + 2*hi + {0,1}
        const float* arow = &sm_code[(code0 + l15) * VQ_D + hi * 2];
        #pragma unroll
        for (int kc = 0; kc < 16; ++kc) {
            v2f av = *(const v2f*)(arow + kc * 4);
            a0 = __builtin_amdgcn_wmma_f32_16x16x4_f32(
                false, av, false, xa0[kc], (short)0, a0, false, false);
            a1 = __builtin_amdgcn_wmma_f32_16x16x4_f32(
                false, av, false, xa1[kc], (short)0, a1, false, false);
        }
        // D layout: vgpr j, this lane -> code = code0 + 8*hi + j, row = l15
        const float* en = &sm_en[code0 + hi * 8];
        #pragma unroll
        for (int j = 0; j < 8; ++j) {
            float e  = en[j];
            int code = code0 + hi * 8 + j;
            float d0 = e - 2.0f * a0[j];
            if (d0 < mv0) { mv0 = d0; mi0 = code; }
            float d1 = e - 2.0f * a1[j];
            if (d1 < mv1) { mv1 = d1; mi1 = code; }
        }
    }
    // combine the two lanes holding the same row (low/high code halves)
    {
        float ov = __shfl_xor(mv0, 16, 32);
        int   oi = __shfl_xor(mi0, 16, 32);
        if (ov < mv0 || (ov == mv0 && oi < mi0)) { mv0 = ov; mi0 = oi; }
    }
    {
        float ov = __shfl_xor(mv1, 16, 32);
        int   oi = __shfl_xor(mi1, 16, 32);
        if (ov < mv1 || (ov == mv1 && oi < mi1)) { mv1 = ov; mi1 = oi; }
    }

    // ---- indices (as float, per harness output dtype) ----
    if (lane < 16) {
        out[OUT_IDX + n0 + lane]      = (float)mi0;
        out[OUT_IDX + n0 + 16 + lane] = (float)mi1;
    }

    // ---- quantized output [B,C,H,W] + commitment-loss SSE ----
    float lsum = 0.0f;
    float* q0 = out + OUT_Q + (((size_t)(b * 64) * 64 + h) * 64 + w0 + l15);
    float* q1 = q0 + 16;
    const float* c0 = &sm_code[mi0 * VQ_D + hi * 2];
    const float* c1 = &sm_code[mi1 * VQ_D + hi * 2];
    #pragma unroll
    for (int kc = 0; kc < 16; ++kc) {
        size_t off = (size_t)(kc * 4 + hi * 2) * VQ_HW;
        v2f e0 = *(const v2f*)(c0 + kc * 4);
        v2f e1 = *(const v2f*)(c1 + kc * 4);
        q0[off]         = e0.x;
        q0[off + VQ_HW] = e0.y;
        q1[off]         = e1.x;
        q1[off + VQ_HW] = e1.y;
        float d;
        d = e0.x - xa0[kc].x; lsum += d * d;
        d = e0.y - xa0[kc].y; lsum += d * d;
        d = e1.x - xa1[kc].x; lsum += d * d;
        d = e1.y - xa1[kc].y; lsum += d * d;
    }

    // ---- histogram + SSE reductions ----
    if (lane < 16) {
        unsafeAtomicAdd(&ws[1 + mi0], 1.0f);
        unsafeAtomicAdd(&ws[1 + mi1], 1.0f);
    }
    atomicAdd(sm_sum, lsum);               // ds_add_f32
    __syncthreads();
    if (tid == 0) unsafeAtomicAdd(&ws[0], sm_sum[0]);
}

__global__ void vq_finalize(float* __restrict__ out, const float* __restrict__ ws) {
    __shared__ float red[VQ_K];
    int t = threadIdx.x;
    float c = ws[1 + t];
    float p = c * (1.0f / (float)VQ_N);
    red[t] = p * __logf(p + 1e-10f);
    __syncthreads();
    for (int s = VQ_K / 2; s > 0; s >>= 1) {
        if (t < s) red[t] += red[t + s];
        __syncthreads();
    }
    if (t == 0) {
        out[OUT_PERP] = __expf(-red[0]);
        out[OUT_LOSS] = 0.25f * ws[0] * (1.0f / ((float)VQ_N * (float)VQ_D));
    }
}

extern "C" void kernel_launch(void* const* d_in, const int* in_sizes, int n_in,
                              void* d_out, int out_size, void* d_ws, size_t ws_size,
                              hipStream_t stream) {
    const float* in = (const float*)d_in[0];   // [32,64,64,64] fp32
    const float* ew = (const float*)d_in[1];   // [512,64] fp32
    float* out = (float*)d_out;
    float* ws  = (float*)d_ws;

    vq_init_ws<<<1, 1024, 0, stream>>>(ws);

    size_t smem = (size_t)(VQ_K * VQ_D + VQ_K + 1) * sizeof(float);  // ~133 KB of 320 KB WGP LDS
    vq_main<<<VQ_N / (32 * 8), 256, smem, stream>>>(in, ew, out, ws);

    vq_finalize<<<1, VQ_K, 0, stream>>>(out, ws);
}